// GraphClassifier_55594056679527
// MI455X (gfx1250) — compile-verified
//
#include <hip/hip_runtime.h>
#include <hip/hip_bf16.h>

typedef __attribute__((ext_vector_type(2))) float v2f;
typedef __attribute__((ext_vector_type(8))) float v8f;

// ---------------- utility kernels ----------------

__global__ void zero_f32(float* __restrict__ p, long long n) {
  long long i = (long long)blockIdx.x * blockDim.x + threadIdx.x;
  if (i < n) p[i] = 0.0f;
}

__global__ void deg_kernel(const int* __restrict__ dst, float* __restrict__ deg, int E) {
  int e = blockIdx.x * blockDim.x + threadIdx.x;
  if (e < E) atomicAdd(&deg[dst[e]], 1.0f);
}

__global__ void dinv_kernel(const float* __restrict__ deg, float* __restrict__ dinv, int N) {
  int n = blockIdx.x * blockDim.x + threadIdx.x;
  if (n < N) dinv[n] = rsqrtf(deg[n] + 1.0f);
}

__global__ void norm_kernel(const int* __restrict__ src, const int* __restrict__ dst,
                            const float* __restrict__ dinv, float* __restrict__ nrm, int E) {
  int e = blockIdx.x * blockDim.x + threadIdx.x;
  if (e < E) nrm[e] = dinv[src[e]] * dinv[dst[e]];
}

// ---------------- fp32 WMMA GEMM: Out[rows,Cout] = A[rows,CIN] @ W[CIN,Cout] ----------------
// Requirements: rows % 16 == 0, CIN in {64,128}, Cout % 64 == 0.
// Block = 4 waves; block computes a 16-row x 64-col output macro-tile.
// A tile (16 x CIN) and B tile (CIN x 64, stored transposed) staged in LDS via coalesced
// b128 global loads; fully-unrolled inner chain: ds_load_2addr_b64 pairs + WMMA.
// WMMA operand striping (wave32):
//   A (16x4 MxK): lanes 0-15 K={0,1}, lanes 16-31 K={2,3}, M = lane&15.
//   B (4x16 KxN): lanes 0-15 K={0,1}, lanes 16-31 K={2,3}, N = lane&15.
//   C (16x16):    VGPR v = row v (lanes 0-15) / row v+8 (lanes 16-31).
#define A_STRIDE 132  // pad: keeps 16B alignment for staging, bank stride 4
#define B_STRIDE 130  // pad: keeps 8B alignment, bank stride 2

template <int CIN>
__global__ void __launch_bounds__(128)
gemm_wmma_f32(const float* __restrict__ A, const float* __restrict__ W,
              float* __restrict__ Out, int Cout) {
  __shared__ float As[16 * A_STRIDE];   // [row_local][k]
  __shared__ float Bs[64 * B_STRIDE];   // [col_local][k]  (transposed W tile)

  const int lane = threadIdx.x;                 // 0..31
  const int half = lane >> 4;                   // 0/1
  const int lm   = lane & 15;
  const int tid  = threadIdx.y * 32 + lane;     // 0..127
  const int tile_r   = blockIdx.x * 16;
  const int col_base = blockIdx.y * 64;

  // ---- stage A tile: 16 x CIN, float4 coalesced loads, b128 LDS stores ----
#pragma unroll
  for (int idx = 0; idx < 16 * CIN; idx += 128 * 4) {
    int i = idx + tid * 4;
    int r = i / CIN, k = i % CIN;               // CIN % 4 == 0 -> chunk stays in row
    float4 v = *(const float4*)(A + (size_t)(tile_r + r) * CIN + k);
    *(float4*)&As[r * A_STRIDE + k] = v;        // A_STRIDE % 4 == 0 -> 16B aligned
  }
  // ---- stage B tile transposed: Bs[c][k] = W[k][col_base + c] ----
#pragma unroll
  for (int idx = 0; idx < CIN * 64; idx += 128 * 4) {
    int i = idx + tid * 4;
    int k = i >> 6, c = i & 63;                 // 4 consecutive cols per thread
    float4 v = *(const float4*)(W + (size_t)k * Cout + col_base + c);
    Bs[(c + 0) * B_STRIDE + k] = v.x;
    Bs[(c + 1) * B_STRIDE + k] = v.y;
    Bs[(c + 2) * B_STRIDE + k] = v.z;
    Bs[(c + 3) * B_STRIDE + k] = v.w;
  }
  __syncthreads();

  const int arow = lm * A_STRIDE + 2 * half;                      // dword base for A
  const int brow = (threadIdx.y * 16 + lm) * B_STRIDE + 2 * half; // dword base for B

  v8f acc = {};
#pragma unroll
  for (int t = 0; t < (CIN >> 2); ++t) {
    v2f a = *(const v2f*)&As[arow + 4 * t];     // ds_load b64 (pairs -> 2addr)
    v2f b = *(const v2f*)&Bs[brow + 4 * t];
    // 8 args: (neg_a, A, neg_b, B, c_mod, C, reuse_a, reuse_b)
    acc = __builtin_amdgcn_wmma_f32_16x16x4_f32(false, a, false, b, (short)0, acc,
                                                false, false);
  }

  const int col = col_base + threadIdx.y * 16 + lm;
  const int rbase = (tile_r + 8 * half) * Cout + col;
#pragma unroll
  for (int v = 0; v < 8; ++v) {
    Out[(size_t)(rbase + v * Cout)] = acc[v];
  }
}

// ---------------- edge scatter: agg[dst] += norm * xw[src], float4 per thread ----------------

__global__ void scatter_kernel(const float* __restrict__ xw, const int* __restrict__ src,
                               const int* __restrict__ dst, const float* __restrict__ nrm,
                               float* __restrict__ agg, int E, int Cout) {
  const int tpe = Cout >> 2;  // threads per edge (float4 each)
  long long tid = (long long)blockIdx.x * blockDim.x + threadIdx.x;
  long long total = (long long)E * tpe;
  if (tid >= total) return;
  int e = (int)(tid / tpe);
  int c = (int)(tid % tpe) * 4;
  int s = src[e], d = dst[e];
  float w = nrm[e];
  const float4 v = *(const float4*)(xw + (size_t)s * Cout + c);
  float* o = agg + (size_t)d * Cout + c;
  atomicAdd(o + 0, w * v.x);
  atomicAdd(o + 1, w * v.y);
  atomicAdd(o + 2, w * v.z);
  atomicAdd(o + 3, w * v.w);
}

// ---------------- finalize (in place): h = relu(agg + dinv^2 * xw + b) ----------------

__global__ void finalize_kernel(float* __restrict__ agg, const float* __restrict__ xw,
                                const float* __restrict__ dinv, const float* __restrict__ bias,
                                int N, int Cout) {
  long long tid = (long long)blockIdx.x * blockDim.x + threadIdx.x;
  if (tid >= (long long)N * Cout) return;
  int n = (int)(tid / Cout);
  int c = (int)(tid % Cout);
  float di = dinv[n];
  float v = agg[tid] + di * di * xw[tid] + bias[c];
  agg[tid] = v > 0.0f ? v : 0.0f;
}

// ---------------- mean pooling ----------------

__global__ void pool_kernel(const float* __restrict__ h, const int* __restrict__ batch,
                            float* __restrict__ sums, float* __restrict__ counts,
                            int N, int C) {
  long long tid = (long long)blockIdx.x * blockDim.x + threadIdx.x;
  if (tid >= (long long)N * C) return;
  int n = (int)(tid / C);
  int c = (int)(tid % C);
  int g = batch[n];
  atomicAdd(&sums[(size_t)g * C + c], h[tid]);
  if (c == 0) atomicAdd(&counts[g], 1.0f);
}

__global__ void emb_kernel(const float* __restrict__ sums, const float* __restrict__ counts,
                           float* __restrict__ emb, int G, int C) {
  int tid = blockIdx.x * blockDim.x + threadIdx.x;
  if (tid >= G * C) return;
  int g = tid / C;
  emb[tid] = sums[tid] / fmaxf(counts[g], 1.0f);
}

// ---------------- classifier tail ----------------

__global__ void bias_relu_kernel(float* __restrict__ z, const float* __restrict__ b,
                                 int rows, int C) {
  int tid = blockIdx.x * blockDim.x + threadIdx.x;
  if (tid >= rows * C) return;
  float v = z[tid] + b[tid % C];
  z[tid] = v > 0.0f ? v : 0.0f;
}

__global__ void head_kernel(const float* __restrict__ z, const float* __restrict__ LW2,
                            const float* __restrict__ Lb2, float* __restrict__ probs,
                            int G, int C) {
  int g = blockIdx.x * blockDim.x + threadIdx.x;
  if (g >= G) return;
  float acc = Lb2[0];
  for (int c = 0; c < C; ++c) acc += z[(size_t)g * C + c] * LW2[c];
  probs[g] = 1.0f / (1.0f + expf(-acc));
}

// ---------------- launcher ----------------

extern "C" void kernel_launch(void* const* d_in, const int* in_sizes, int n_in,
                              void* d_out, int out_size, void* d_ws, size_t ws_size,
                              hipStream_t stream) {
  const float* x    = (const float*)d_in[0];
  const int*   ei   = (const int*)d_in[1];
  const int*   batch= (const int*)d_in[2];
  const float* W1   = (const float*)d_in[3];
  const float* b1   = (const float*)d_in[4];
  const float* W2   = (const float*)d_in[5];
  const float* b2   = (const float*)d_in[6];
  const float* W3   = (const float*)d_in[7];
  const float* b3   = (const float*)d_in[8];
  const float* LW1  = (const float*)d_in[9];
  const float* Lb1  = (const float*)d_in[10];
  const float* LW2  = (const float*)d_in[11];
  const float* Lb2  = (const float*)d_in[12];

  const int N = in_sizes[0] / 128;     // 50000 nodes (multiple of 16)
  const int E = in_sizes[1] / 2;       // 600000 edges
  const int G = out_size / 65;         // 128 graphs

  const int* src = ei;
  const int* dst = ei + E;

  // workspace carve-up (floats)
  float* ws = (float*)d_ws;
  size_t off = 0;
  float* bufA   = ws + off; off += (size_t)N * 128;   // xw scratch
  float* bufB   = ws + off; off += (size_t)N * 128;   // agg / h scratch
  float* deg    = ws + off; off += (size_t)N;
  float* dinv   = ws + off; off += (size_t)N;
  float* nrm    = ws + off; off += (size_t)E;
  float* sums   = ws + off; off += (size_t)G * 64;
  float* counts = ws + off; off += (size_t)G;         // contiguous after sums
  float* zbuf   = ws + off; off += (size_t)G * 128;
  (void)ws_size; (void)n_in;

  float* probs = (float*)d_out;        // [G]
  float* emb   = (float*)d_out + G;    // [G,64]

  const int TB = 256;
  auto blks = [](long long n, int tb) { return (unsigned)((n + tb - 1) / tb); };

  // ---- degree / normalization (fixed graph structure, shared by all layers) ----
  zero_f32<<<blks(N, TB), TB, 0, stream>>>(deg, N);
  deg_kernel<<<blks(E, TB), TB, 0, stream>>>(dst, deg, E);
  dinv_kernel<<<blks(N, TB), TB, 0, stream>>>(deg, dinv, N);
  norm_kernel<<<blks(E, TB), TB, 0, stream>>>(src, dst, dinv, nrm, E);

  // ---- one GCN layer ----
  auto layer = [&](const float* Xin, const float* Wt, const float* bias,
                   int Cin, int Cout) {
    dim3 gb(32, 4);
    dim3 gg(N / 16, (Cout + 63) / 64);
    if (Cin == 128)
      gemm_wmma_f32<128><<<gg, gb, 0, stream>>>(Xin, Wt, bufA, Cout);
    else
      gemm_wmma_f32<64><<<gg, gb, 0, stream>>>(Xin, Wt, bufA, Cout);
    long long nc = (long long)N * Cout;
    zero_f32<<<blks(nc, TB), TB, 0, stream>>>(bufB, nc);
    long long tot = (long long)E * (Cout / 4);
    scatter_kernel<<<blks(tot, TB), TB, 0, stream>>>(bufA, src, dst, nrm, bufB, E, Cout);
    finalize_kernel<<<blks(nc, TB), TB, 0, stream>>>(bufB, bufA, dinv, bias, N, Cout);
  };

  layer(x,    W1, b1, 128, 128);   // h1 -> bufB
  layer(bufB, W2, b2, 128, 128);   // h2 -> bufB
  layer(bufB, W3, b3, 128, 64);    // h3 -> bufB [N,64]

  // ---- global mean pool ----
  zero_f32<<<blks((long long)G * 65, TB), TB, 0, stream>>>(sums, (long long)G * 65);
  pool_kernel<<<blks((long long)N * 64, TB), TB, 0, stream>>>(bufB, batch, sums, counts, N, 64);
  emb_kernel<<<blks((long long)G * 64, TB), TB, 0, stream>>>(sums, counts, emb, G, 64);

  // ---- classifier: z = relu(emb @ LW1 + Lb1); probs = sigmoid(z @ LW2 + Lb2) ----
  {
    dim3 gb(32, 4);
    dim3 gg(G / 16, 2);
    gemm_wmma_f32<64><<<gg, gb, 0, stream>>>(emb, LW1, zbuf, 128);
  }
  bias_relu_kernel<<<blks((long long)G * 128, TB), TB, 0, stream>>>(zbuf, Lb1, G, 128);
  head_kernel<<<blks(G, 128), 128, 0, stream>>>(zbuf, LW2, Lb2, probs, G, 128);
}